// QuantizerEMA_85624468013074
// MI455X (gfx1250) — compile-verified
//
#include <hip/hip_runtime.h>
#include <hip/hip_bf16.h>

typedef __attribute__((ext_vector_type(2))) float v2f;
typedef __attribute__((ext_vector_type(8))) float v8f;
typedef __attribute__((ext_vector_type(4))) unsigned int v4u;
typedef __attribute__((ext_vector_type(8))) int v8i;
typedef __attribute__((ext_vector_type(4))) int v4i;

#define DECAY  0.99f
#define COMMIT 0.25f
#define EPSL   1e-5f
#define B_   32
#define HH   64
#define WW_  64
#define DD   64
#define KK   512
#define HW   (HH * WW_)          // 4096
#define NTOK (B_ * HW)           // 131072

// workspace float layout
#define WS_CLOSEST 0                       // int[NTOK]
#define WS_NI      NTOK                    // float[KK]
#define WS_DW      (NTOK + KK)             // float[KK*DD]
#define WS_LOSS    (NTOK + KK + KK * DD)   // float[1]

// output float layout (tuple concatenated flat, return order)
#define OUT_Q    0
#define OUT_IDX  (B_ * DD * HW)            // 8388608
#define OUT_LOSS (OUT_IDX + NTOK)          // 8519680
#define OUT_EMB  (OUT_LOSS + 1)
#define OUT_EMA  (OUT_EMB + KK * DD)
#define OUT_CS   (OUT_EMA + KK * DD)

#define SE_STRIDE 68                       // 64 + 4 pad floats (bank-conflict kill)

// ---------------------------------------------------------------- zero accum
__global__ void k_zero(float* ws) {
    int i = blockIdx.x * blockDim.x + threadIdx.x;
    const int n = KK + KK * DD + 1;
    if (i < n) ws[WS_NI + i] = 0.0f;
}

// --------------------------------------------------- WMMA distance + argmin
// score(m,c) = z_m . e_c - 0.5*||e_c||^2   (argmax == argmin of distance)
// 128 threads = 4 waves; each wave owns 32 tokens (two 16-row A tiles with
// independent accumulators interleaved on the same B fragment).
__global__ void k_argmin(const float* __restrict__ z,
                         const float* __restrict__ emb,
                         int* __restrict__ closest,
                         float* __restrict__ outIdx) {
    extern __shared__ float smem[];
    float* sE    = smem;                   // [KK][SE_STRIDE], starts at LDS offset 0
    float* sNorm = smem + KK * SE_STRIDE;  // [KK] holds -0.5*||e||^2

    const int tid = threadIdx.x;

    // ---- TDM: async tensor DMA emb[512][64] f32 -> LDS, padded 4 DWORDs per
    // 256-byte row => LDS row stride = 68 floats. Wave 0 issues + waits.
    if (tid < 32) {
        unsigned long long ga = (unsigned long long)(uintptr_t)emb;
        v4u g0;
        g0[0] = 1u;                                   // count=1, normal mode
        g0[1] = 0u;                                   // lds_addr = 0 (dyn smem base)
        g0[2] = (unsigned int)(ga & 0xFFFFFFFFull);   // global_addr[31:0]
        g0[3] = (unsigned int)((ga >> 32) & 0x01FFFFFFull) | (2u << 30); // [56:32] | type=2
        v8i g1;
        g1[0] = (2 << 16)        // data_size = 4 bytes
              | (1 << 20)        // pad_enable
              | (5 << 22)        // pad_interval: 2^(5+1)=64 DWORDs = 256 B
              | (3 << 25);       // pad_amount: 4 DWORDs = 16 B
        g1[1] = (DD << 16);      // tensor_dim0 = 64   (bits 79:48, low half)
        g1[2] = (KK << 16);      // tensor_dim1 = 512  (bits 111:80, low half)
        g1[3] = (DD << 16);      // tile_dim0 = 64     (bits 127:112)
        g1[4] = KK;              // tile_dim1 = 512    (bits 143:128)
        g1[5] = DD;              // tensor_dim0_stride = 64 (elements)
        g1[6] = 0;
        g1[7] = 0;
        v4i zero4 = {0, 0, 0, 0};
        v8i zero8 = {0, 0, 0, 0, 0, 0, 0, 0};
        __builtin_amdgcn_tensor_load_to_lds(g0, g1, zero4, zero4, zero8, 0);
        __builtin_amdgcn_s_wait_tensorcnt((short)0);
    }
    __syncthreads();

    // per-code -0.5*||e||^2
    for (int k = tid; k < KK; k += 128) {
        float s = 0.0f;
        #pragma unroll
        for (int d = 0; d < DD; ++d) { float e = sE[k * SE_STRIDE + d]; s += e * e; }
        sNorm[k] = -0.5f * s;
    }
    __syncthreads();

    const int wave = tid >> 5;
    const int lane = tid & 31;
    const int half = lane >> 4;            // 0: K={0,1}, 1: K={2,3} per 16x4 A layout
    const int l16  = lane & 15;
    const int t0   = (blockIdx.x * 4 + wave) * 32;   // this wave's 32 tokens

    // preload two A tile sets (rows t0+l16 and t0+16+l16), 16 k-steps of depth 4
    v2f a0[16], a1[16];
    const float* zr0 = z + (size_t)(t0 + l16) * DD + 2 * half;
    const float* zr1 = z + (size_t)(t0 + 16 + l16) * DD + 2 * half;
    #pragma unroll
    for (int s = 0; s < 16; ++s) {
        a0[s] = *reinterpret_cast<const v2f*>(zr0 + 4 * s);
        a1[s] = *reinterpret_cast<const v2f*>(zr1 + 4 * s);
    }

    float bv0[8], bv1[8];
    int   bi0[8], bi1[8];
    #pragma unroll
    for (int v = 0; v < 8; ++v) {
        bv0[v] = -3.4e38f; bi0[v] = 0;
        bv1[v] = -3.4e38f; bi1[v] = 0;
    }

    for (int ct = 0; ct < KK / 16; ++ct) {
        const int c = ct * 16 + l16;       // code column owned by this lane
        const float nh = sNorm[c];
        v8f acc0, acc1;
        #pragma unroll
        for (int v = 0; v < 8; ++v) { acc0[v] = nh; acc1[v] = nh; }

        const float* erow = sE + c * SE_STRIDE + 2 * half;
        #pragma unroll
        for (int s = 0; s < 16; ++s) {
            v2f b = *reinterpret_cast<const v2f*>(erow + 4 * s);
            acc0 = __builtin_amdgcn_wmma_f32_16x16x4_f32(
                false, a0[s], false, b, (short)0, acc0, false, false);
            acc1 = __builtin_amdgcn_wmma_f32_16x16x4_f32(
                false, a1[s], false, b, (short)0, acc1, false, false);
        }
        #pragma unroll
        for (int v = 0; v < 8; ++v) {
            if (acc0[v] > bv0[v]) { bv0[v] = acc0[v]; bi0[v] = c; }
            if (acc1[v] > bv1[v]) { bv1[v] = acc1[v]; bi1[v] = c; }
        }
    }

    // butterfly reduce across the 16 lanes of each half-wave (C layout:
    // lanes 0-15 hold rows 0-7 in vgpr v, lanes 16-31 hold rows 8-15)
    #pragma unroll
    for (int off = 1; off < 16; off <<= 1) {
        #pragma unroll
        for (int v = 0; v < 8; ++v) {
            float ov0 = __shfl_xor(bv0[v], off, 32);
            int   oi0 = __shfl_xor(bi0[v], off, 32);
            if (ov0 > bv0[v] || (ov0 == bv0[v] && oi0 < bi0[v])) { bv0[v] = ov0; bi0[v] = oi0; }
            float ov1 = __shfl_xor(bv1[v], off, 32);
            int   oi1 = __shfl_xor(bi1[v], off, 32);
            if (ov1 > bv1[v] || (ov1 == bv1[v] && oi1 < bi1[v])) { bv1[v] = ov1; bi1[v] = oi1; }
        }
    }
    if (l16 == 0) {
        #pragma unroll
        for (int v = 0; v < 8; ++v) {
            int tok0 = t0 + v + 8 * half;
            closest[tok0] = bi0[v];
            outIdx[tok0]  = (float)bi0[v];
            int tok1 = t0 + 16 + v + 8 * half;
            closest[tok1] = bi1[v];
            outIdx[tok1]  = (float)bi1[v];
        }
    }
}

// ------------------------- gather + NCHW transpose + loss + EMA scatter accum
__global__ void k_gather(const float* __restrict__ z,
                         const float* __restrict__ emb,
                         const int* __restrict__ closest,
                         float* __restrict__ outQ,
                         float* __restrict__ nI,
                         float* __restrict__ dw,
                         float* __restrict__ lossAcc) {
    __shared__ float tile[32][33];
    const int bid = blockIdx.x;            // B_ * (HW/32) * (DD/32) blocks
    const int dt  = bid & 1;               // DD/32 = 2
    const int hwt = (bid >> 1) & 127;      // HW/32 = 128
    const int b   = bid >> 8;
    const int tx  = threadIdx.x & 31;
    const int ty  = threadIdx.x >> 5;      // 0..7
    const int d0  = dt * 32, hw0 = hwt * 32;

    float lsum = 0.0f;
    #pragma unroll
    for (int i = 0; i < 4; ++i) {
        int r   = ty + 8 * i;
        int tok = b * HW + hw0 + r;
        int idx = closest[tok];
        int d   = d0 + tx;
        float zv = z[(size_t)tok * DD + d];
        float ev = emb[idx * DD + d];
        tile[r][tx] = ev;
        float df = ev - zv;
        lsum += df * df;
        atomicAdd(&dw[idx * DD + d], zv);
        if (d == 0) atomicAdd(&nI[idx], 1.0f);
    }
    __syncthreads();
    #pragma unroll
    for (int i = 0; i < 4; ++i) {
        int dd = ty + 8 * i;
        outQ[(size_t)(b * DD + d0 + dd) * HW + hw0 + tx] = tile[tx][dd];
    }
    // wave-level loss reduction, one atomic per wave
    #pragma unroll
    for (int off = 16; off > 0; off >>= 1) lsum += __shfl_down(lsum, off, 32);
    if ((threadIdx.x & 31) == 0) atomicAdd(lossAcc, lsum);
}

// ----------------------------------------------- EMA finalize (cluster sizes)
__global__ void k_ema1(const float* __restrict__ clusterSize,
                       const float* __restrict__ nI,
                       const float* __restrict__ lossAcc,
                       float* __restrict__ outCS,
                       float* __restrict__ outLoss) {
    __shared__ float red[KK];
    const int k = threadIdx.x;             // KK threads, 1 block
    float cs = clusterSize[k] * DECAY + nI[k] * (1.0f - DECAY);
    red[k] = cs;
    __syncthreads();
    for (int s = KK / 2; s > 0; s >>= 1) {
        if (k < s) red[k] += red[k + s];
        __syncthreads();
    }
    float n = red[0];
    outCS[k] = (cs + EPSL) / (n + KK * EPSL) * n;
    if (k == 0) outLoss[0] = lossAcc[0] * (COMMIT / (float)(NTOK * DD));
}

// ----------------------------------------------- EMA finalize (embeddings)
__global__ void k_ema2(const float* __restrict__ ema,
                       const float* __restrict__ dw,
                       const float* __restrict__ outCS,
                       float* __restrict__ outEma,
                       float* __restrict__ outEmb) {
    int i = blockIdx.x * blockDim.x + threadIdx.x;
    if (i < KK * DD) {
        float nev = ema[i] * DECAY + dw[i] * (1.0f - DECAY);
        outEma[i] = nev;
        outEmb[i] = nev / outCS[i >> 6];   // DD = 64
    }
}

extern "C" void kernel_launch(void* const* d_in, const int* in_sizes, int n_in,
                              void* d_out, int out_size, void* d_ws, size_t ws_size,
                              hipStream_t stream) {
    const float* z   = (const float*)d_in[0];
    const float* emb = (const float*)d_in[1];
    const float* ema = (const float*)d_in[2];
    const float* cs  = (const float*)d_in[3];
    float* out = (float*)d_out;
    float* ws  = (float*)d_ws;
    int*   closest = (int*)d_ws;           // WS_CLOSEST

    float* nI      = ws + WS_NI;
    float* dw      = ws + WS_DW;
    float* lossAcc = ws + WS_LOSS;

    // 1) zero scatter accumulators (re-run every call; deterministic)
    const int nz = KK + KK * DD + 1;
    k_zero<<<(nz + 255) / 256, 256, 0, stream>>>(ws);

    // 2) fused distance-GEMM + argmin (TDM-staged LDS + WMMA f32 16x16x4)
    const size_t smemBytes = (size_t)KK * SE_STRIDE * sizeof(float) + KK * sizeof(float);
    (void)hipFuncSetAttribute((const void*)k_argmin,
                              hipFuncAttributeMaxDynamicSharedMemorySize, (int)smemBytes);
    k_argmin<<<NTOK / 128, 128, smemBytes, stream>>>(z, emb, closest, out + OUT_IDX);

    // 3) gather + transpose + loss + dw/n_i scatter
    k_gather<<<B_ * (HW / 32) * (DD / 32), 256, 0, stream>>>(
        z, emb, closest, out + OUT_Q, nI, dw, lossAcc);

    // 4) EMA finalize
    k_ema1<<<1, KK, 0, stream>>>(cs, nI, lossAcc, out + OUT_CS, out + OUT_LOSS);
    k_ema2<<<(KK * DD + 255) / 256, 256, 0, stream>>>(
        ema, dw, out + OUT_CS, out + OUT_EMA, out + OUT_EMB);
}